// FullModel_39934605918856
// MI455X (gfx1250) — compile-verified
//
#include <hip/hip_runtime.h>

typedef float v2f __attribute__((ext_vector_type(2)));
typedef float v8f __attribute__((ext_vector_type(8)));

#define BATCH 64
#define TT    12
#define NNODE 716
#define DD    64
#define EE    32
#define QB    (TT * NNODE)   /* 8592 rows per batch, divisible by 16 */
#define TILES (QB / 16)      /* 537 */
#define WPB   4              /* waves per block in the fused kernel  */

__device__ __forceinline__ v8f wmma_f32(v2f a, v2f b, v8f c) {
  // V_WMMA_F32_16X16X4_F32 : D = A(16x4 f32) * B(4x16 f32) + C(16x16 f32)
  return __builtin_amdgcn_wmma_f32_16x16x4_f32(false, a, false, b,
                                               (short)0, c, false, false);
}

// ---------------------------------------------------------------------------
// Zero a float buffer (hf accumulator).
// ---------------------------------------------------------------------------
__global__ void zero_kernel(float* __restrict__ p, int n) {
  int i = blockIdx.x * blockDim.x + threadIdx.x;
  if (i < n) p[i] = 0.f;
}

// ---------------------------------------------------------------------------
// hf[b,e,d] = sum_q softmax(x[b,q,:]@EC)[e] * x[b,q,d]
// One lane per edge e (E == 32 == wave32). 64 waves per batch, atomic reduce.
// ---------------------------------------------------------------------------
__global__ __launch_bounds__(256) void hf_kernel(const float* __restrict__ X,
                                                 const float* __restrict__ EC,
                                                 float* __restrict__ HF) {
  __shared__ float sEC[DD * EE];
  const int tid = threadIdx.x;
  for (int i = tid; i < DD * EE; i += 256) sEC[i] = EC[i];
  __syncthreads();

  const int b    = blockIdx.y;
  const int lane = tid & 31;                       // edge index
  const int wg   = blockIdx.x * 8 + (tid >> 5);    // 0..63 waves per batch
  const float* Xb = X + (size_t)b * QB * DD;

  float acc[DD];
#pragma unroll
  for (int d = 0; d < DD; ++d) acc[d] = 0.f;

  for (int q = wg; q < QB; q += 64) {
    const float* xr = Xb + (size_t)q * DD;
    float lg = 0.f;
#pragma unroll
    for (int d = 0; d < DD; ++d) lg += xr[d] * sEC[d * EE + lane];
    // softmax across the 32 lanes
    float mx = lg;
#pragma unroll
    for (int off = 16; off > 0; off >>= 1) mx = fmaxf(mx, __shfl_xor(mx, off, 32));
    float ev = __expf(lg - mx);
    float sm = ev;
#pragma unroll
    for (int off = 16; off > 0; off >>= 1) sm += __shfl_xor(sm, off, 32);
    float a = ev / sm;
#pragma unroll
    for (int d = 0; d < DD; ++d) acc[d] += a * xr[d];
  }

  float* dst = HF + ((size_t)b * EE + lane) * DD;
#pragma unroll
  for (int d = 0; d < DD; ++d) atomicAdd(&dst[d], acc[d]);
}

// ---------------------------------------------------------------------------
// ho[b,e,d] = relu(sum_f EM[e,f]*hf[b,f,d]) + hf[b,e,d]
// ---------------------------------------------------------------------------
__global__ __launch_bounds__(256) void ho_kernel(const float* __restrict__ EM,
                                                 const float* __restrict__ HF,
                                                 float* __restrict__ HO) {
  __shared__ float sEM[EE * EE];
  const int tid = threadIdx.x;
  for (int i = tid; i < EE * EE; i += 256) sEM[i] = EM[i];
  __syncthreads();

  const int b = blockIdx.x;
  const float* hfb = HF + (size_t)b * EE * DD;
  float* hob = HO + (size_t)b * EE * DD;
  for (int o = tid; o < EE * DD; o += 256) {
    int e = o >> 6, d = o & 63;
    float acc = 0.f;
#pragma unroll
    for (int f = 0; f < EE; ++f) acc += sEM[e * EE + f] * hfb[f * DD + d];
    float r = acc > 0.f ? acc : 0.f;
    hob[o] = r + hfb[o];
  }
}

// ---------------------------------------------------------------------------
// Fused STGCN + hyper-back-projection + both LayerNorms + 0.5*(out1+out2).
// One wave per 16-row tile of the flattened (t,n) axis. All GEMMs via
// V_WMMA_F32_16X16X4_F32 (exact fp32).
// ---------------------------------------------------------------------------
__global__ __launch_bounds__(WPB * 32) void stgcn_hyper_kernel(
    const float* __restrict__ X,
    const float* __restrict__ W1, const float* __restrict__ B1,
    const float* __restrict__ W2, const float* __restrict__ B2,
    const float* __restrict__ G1, const float* __restrict__ Bl1,
    const float* __restrict__ EC, const float* __restrict__ HO,
    const float* __restrict__ G2, const float* __restrict__ Bl2,
    float* __restrict__ Out) {
  __shared__ float sAb[WPB][16 * 64];   // Abar = 0.5*(x[t]+x[t-1]) tile
  __shared__ float sS [WPB][16 * 64];   // full + x (pre-LN1)
  __shared__ float sS2[WPB][16 * 64];   // relu(y) + x (pre-LN2)
  __shared__ float sL [WPB][16 * 32];   // logits -> assign
  __shared__ float sStat[WPB][16 * 4];  // mu1, rs1, mu2, rs2 per row

  const int tid  = threadIdx.x;
  const int wave = tid >> 5;
  const int lane = tid & 31;
  const int b    = blockIdx.y;
  const int tile = blockIdx.x * WPB + wave;
  if (tile >= TILES) return;            // wave-uniform

  const int qbase = tile * 16;
  const float* Xb  = X + (size_t)b * QB * DD;
  const float* hob = HO + (size_t)b * EE * DD;
  const size_t rowbase = ((size_t)b * QB + qbase) * DD;

  // stage Abar (coalesced global reads)
  for (int i = lane; i < 16 * 64; i += 32) {
    int r = i >> 6, c = i & 63;
    int q = qbase + r;
    float cur = Xb[(size_t)q * DD + c];
    float prv = (q >= NNODE) ? Xb[(size_t)(q - NNODE) * DD + c] : 0.f;
    sAb[wave][i] = 0.5f * (cur + prv);
  }
  __builtin_amdgcn_wave_barrier();

  const int m = lane & 15;      // A-row / B-col / C-col within tile
  const int h = lane >> 4;      // K-half select

  // ---- g1 = Abar@W1 + b1 ; g2 = Abar@W2 + b2 ; full = relu(g1*g2)+g1 ----
  for (int j = 0; j < 4; ++j) {
    float bb1 = B1[16 * j + m];
    float bb2 = B2[16 * j + m];
    v8f a1, a2;
#pragma unroll
    for (int r = 0; r < 8; ++r) { a1[r] = bb1; a2[r] = bb2; }
#pragma unroll
    for (int k = 0; k < 16; ++k) {
      int kk = 4 * k + 2 * h;
      v2f af;  af.x  = sAb[wave][m * 64 + kk];      af.y  = sAb[wave][m * 64 + kk + 1];
      v2f bf1; bf1.x = W1[kk * 64 + 16 * j + m];    bf1.y = W1[(kk + 1) * 64 + 16 * j + m];
      v2f bf2; bf2.x = W2[kk * 64 + 16 * j + m];    bf2.y = W2[(kk + 1) * 64 + 16 * j + m];
      a1 = wmma_f32(af, bf1, a1);
      a2 = wmma_f32(af, bf2, a2);
    }
#pragma unroll
    for (int r = 0; r < 8; ++r) {
      float g1v = a1[r], g2v = a2[r];
      float fu = g1v * g2v; fu = fu > 0.f ? fu : 0.f; fu += g1v;
      int row = r + 8 * h;
      float xc = Xb[(size_t)(qbase + row) * DD + 16 * j + m];
      sS[wave][row * 64 + 16 * j + m] = fu + xc;
    }
  }

  // ---- logits = x@EC (16x32), then row softmax in LDS ----
  for (int j = 0; j < 2; ++j) {
    v8f lg = {};
#pragma unroll
    for (int k = 0; k < 16; ++k) {
      int kk = 4 * k + 2 * h;
      v2f af; af.x = Xb[(size_t)(qbase + m) * DD + kk];
              af.y = Xb[(size_t)(qbase + m) * DD + kk + 1];
      v2f bf; bf.x = EC[kk * EE + 16 * j + m];
              bf.y = EC[(kk + 1) * EE + 16 * j + m];
      lg = wmma_f32(af, bf, lg);
    }
#pragma unroll
    for (int r = 0; r < 8; ++r) sL[wave][(r + 8 * h) * 32 + 16 * j + m] = lg[r];
  }
  __builtin_amdgcn_wave_barrier();
  if (lane < 16) {
    float mx = -3.4e38f;
    for (int e = 0; e < EE; ++e) mx = fmaxf(mx, sL[wave][lane * 32 + e]);
    float sm = 0.f;
    for (int e = 0; e < EE; ++e) {
      float v = __expf(sL[wave][lane * 32 + e] - mx);
      sL[wave][lane * 32 + e] = v; sm += v;
    }
    float inv = 1.f / sm;
    for (int e = 0; e < EE; ++e) sL[wave][lane * 32 + e] *= inv;
  }
  __builtin_amdgcn_wave_barrier();

  // ---- y = relu(assign @ ho[b]) ; s2 = y + x ----
  for (int j = 0; j < 4; ++j) {
    v8f yv = {};
#pragma unroll
    for (int k = 0; k < 8; ++k) {
      int kk = 4 * k + 2 * h;
      v2f af; af.x = sL[wave][m * 32 + kk]; af.y = sL[wave][m * 32 + kk + 1];
      v2f bf; bf.x = hob[kk * 64 + 16 * j + m]; bf.y = hob[(kk + 1) * 64 + 16 * j + m];
      yv = wmma_f32(af, bf, yv);
    }
#pragma unroll
    for (int r = 0; r < 8; ++r) {
      float y = yv[r]; y = y > 0.f ? y : 0.f;
      int row = r + 8 * h;
      float xc = Xb[(size_t)(qbase + row) * DD + 16 * j + m];
      sS2[wave][row * 64 + 16 * j + m] = y + xc;
    }
  }
  __builtin_amdgcn_wave_barrier();

  // ---- LayerNorm stats per row ----
  if (lane < 16) {
    int row = lane;
    float s1 = 0.f, s2 = 0.f;
    for (int c = 0; c < 64; ++c) { s1 += sS[wave][row * 64 + c]; s2 += sS2[wave][row * 64 + c]; }
    float mu1 = s1 * (1.f / 64.f), mu2 = s2 * (1.f / 64.f);
    float v1 = 0.f, v2 = 0.f;
    for (int c = 0; c < 64; ++c) {
      float d1 = sS[wave][row * 64 + c] - mu1;  v1 += d1 * d1;
      float d2 = sS2[wave][row * 64 + c] - mu2; v2 += d2 * d2;
    }
    sStat[wave][row * 4 + 0] = mu1;
    sStat[wave][row * 4 + 1] = rsqrtf(v1 * (1.f / 64.f) + 1e-5f);
    sStat[wave][row * 4 + 2] = mu2;
    sStat[wave][row * 4 + 3] = rsqrtf(v2 * (1.f / 64.f) + 1e-5f);
  }
  __builtin_amdgcn_wave_barrier();

  // ---- out = 0.5*(LN1(s)*g1+b1 + LN2(s2)*g2+b2), coalesced stores ----
  for (int r = 0; r < 16; ++r) {
    float mu1 = sStat[wave][r * 4 + 0], rs1 = sStat[wave][r * 4 + 1];
    float mu2 = sStat[wave][r * 4 + 2], rs2 = sStat[wave][r * 4 + 3];
#pragma unroll
    for (int half = 0; half < 2; ++half) {
      int c = lane + 32 * half;
      float o1 = (sS[wave][r * 64 + c] - mu1) * rs1 * G1[c] + Bl1[c];
      float o2 = (sS2[wave][r * 64 + c] - mu2) * rs2 * G2[c] + Bl2[c];
      Out[rowbase + (size_t)r * 64 + c] = 0.5f * (o1 + o2);
    }
  }
}

// ---------------------------------------------------------------------------
extern "C" void kernel_launch(void* const* d_in, const int* in_sizes, int n_in,
                              void* d_out, int out_size, void* d_ws, size_t ws_size,
                              hipStream_t stream) {
  const float* x0  = (const float*)d_in[0];
  const float* W1  = (const float*)d_in[1];   // [3,64,64]
  const float* b1  = (const float*)d_in[2];   // [3,64]
  const float* W2  = (const float*)d_in[3];
  const float* b2  = (const float*)d_in[4];
  const float* g1  = (const float*)d_in[5];   // [3,64]
  const float* bl1 = (const float*)d_in[6];
  const float* ec  = (const float*)d_in[7];   // [64,32]
  const float* em  = (const float*)d_in[8];   // [32,32]
  const float* g2  = (const float*)d_in[9];   // [64]
  const float* bl2 = (const float*)d_in[10];

  float* out = (float*)d_out;
  float* wsf = (float*)d_ws;

  const size_t XN = (size_t)BATCH * QB * DD;  // 35,192,832 floats
  float* xping = wsf;                          // ping buffer for x
  float* hf    = wsf + XN;                     // [B,32,64]
  float* ho    = hf + (size_t)BATCH * EE * DD; // [B,32,64]
  const int HFN = BATCH * EE * DD;             // 131072

  const float* Xin[3]  = { x0,  out,   xping };
  float*       Xout[3] = { out, xping, out   };

  dim3 mainGrid((TILES + WPB - 1) / WPB, BATCH);  // (135, 64)

  for (int i = 0; i < 3; ++i) {
    zero_kernel<<<(HFN + 255) / 256, 256, 0, stream>>>(hf, HFN);
    hf_kernel<<<dim3(8, BATCH), 256, 0, stream>>>(Xin[i], ec, hf);
    ho_kernel<<<BATCH, 256, 0, stream>>>(em, hf, ho);
    stgcn_hyper_kernel<<<mainGrid, WPB * 32, 0, stream>>>(
        Xin[i],
        W1 + (size_t)i * DD * DD, b1 + (size_t)i * DD,
        W2 + (size_t)i * DD * DD, b2 + (size_t)i * DD,
        g1 + (size_t)i * DD, bl1 + (size_t)i * DD,
        ec, ho, g2, bl2, Xout[i]);
  }
}